// ContinuousAttention_74809740362409
// MI455X (gfx1250) — compile-verified
//
#include <hip/hip_runtime.h>
#include <math.h>

// ---------------- problem constants ----------------
#define BS     32
#define L      1024      // SEQ_LEN
#define HDIM   1024
#define NB     256
#define SQRT2PI_INV 0.3989422804014327f

typedef float  v2f __attribute__((ext_vector_type(2)));
typedef float  v8f __attribute__((ext_vector_type(8)));

__device__ __forceinline__ float posf(int s) { return (s + 0.5f) * (1.0f / 1024.0f); }
__device__ __forceinline__ float mu_of(int k) { return (float)(k >> 1) * (1.0f / 127.0f); }
__device__ __forceinline__ float sig_of(int k) { return (k & 1) ? 0.5f : 0.1f; }

// ---------------- K1: build F (NB x L) ----------------
__global__ void fill_F(float* __restrict__ F) {
    int idx = blockIdx.x * 256 + threadIdx.x;     // 256*1024 elements
    int k = idx >> 10, s = idx & 1023;
    float mu = mu_of(k), sg = sig_of(k);
    float d = posf(s) - mu;
    F[idx] = expf(-(d * d) / (2.0f * sg * sg)) * (SQRT2PI_INV / sg);
}

// ---------------- K2: M = F*F^T (+0.01 I), write augmented [M | I] f64 ----------------
// one wave per 16x16 tile; grid (16,16), block 32
__global__ void gemm_M_wmma(const float* __restrict__ F, double* __restrict__ Aug) {
    const int i0 = blockIdx.y * 16, j0 = blockIdx.x * 16;
    const int lane = threadIdx.x, half = lane >> 4, idx = lane & 15;
    v8f acc = {};
    for (int k = 0; k < L; k += 4) {
        int kk = k + 2 * half;
        v2f a, b;
        a.x = F[(i0 + idx) * L + kk];     a.y = F[(i0 + idx) * L + kk + 1];
        b.x = F[(j0 + idx) * L + kk];     b.y = F[(j0 + idx) * L + kk + 1];
        acc = __builtin_amdgcn_wmma_f32_16x16x4_f32(false, a, false, b, (short)0, acc, false, false);
    }
#pragma unroll
    for (int t = 0; t < 8; ++t) {
        int i = i0 + t + 8 * half, j = j0 + idx;
        Aug[(size_t)i * 512 + j]       = (double)acc[t] + ((i == j) ? 0.01 : 0.0);
        Aug[(size_t)i * 512 + 256 + j] = (i == j) ? 1.0 : 0.0;
    }
}

// ---------------- K3: Gauss-Jordan inverse of M (f64), in augmented form ----------------
// single block, 1024 threads: thread = (row, 128-col chunk)
__global__ void gauss_jordan(double* __restrict__ Aug) {
    __shared__ double prow[512];
    const int tid = threadIdx.x;
    const int row = tid >> 2, c0 = (tid & 3) * 128;
    for (int p = 0; p < 256; ++p) {
        double pivot = Aug[(size_t)p * 512 + p];
        if (tid < 512) prow[tid] = Aug[(size_t)p * 512 + tid] / pivot;
        __syncthreads();
        double f = Aug[(size_t)row * 512 + p];
        __syncthreads();
        if (row == p) {
            for (int j = c0; j < c0 + 128; ++j) Aug[(size_t)row * 512 + j] = prow[j];
        } else {
            for (int j = c0; j < c0 + 128; ++j) Aug[(size_t)row * 512 + j] -= f * prow[j];
        }
        __syncthreads();
    }
}

// ---------------- K4: G = F^T * Minv  (L x NB, f32) ----------------
// grid (64,16) tiles of 16x16, block 32
__global__ void gemm_G_wmma(const float* __restrict__ F, const double* __restrict__ Aug,
                            float* __restrict__ G) {
    const int s0 = blockIdx.x * 16, j0 = blockIdx.y * 16;
    const int lane = threadIdx.x, half = lane >> 4, idx = lane & 15;
    v8f acc = {};
    for (int k = 0; k < NB; k += 4) {
        int kk = k + 2 * half;
        v2f a, b;
        a.x = F[(size_t)kk * L + s0 + idx];
        a.y = F[(size_t)(kk + 1) * L + s0 + idx];
        b.x = (float)Aug[(size_t)kk * 512 + 256 + j0 + idx];
        b.y = (float)Aug[(size_t)(kk + 1) * 512 + 256 + j0 + idx];
        acc = __builtin_amdgcn_wmma_f32_16x16x4_f32(false, a, false, b, (short)0, acc, false, false);
    }
#pragma unroll
    for (int t = 0; t < 8; ++t)
        G[(size_t)(s0 + t + 8 * half) * NB + j0 + idx] = acc[t];
}

// ---------------- K5: scores[b][s] = q[b].k[b][s] / 32 ----------------
// grid (8 s-chunks, 32 batches), block 256 (8 waves, 16 rows per wave)
__global__ void scores_kernel(const float* __restrict__ q, const float* __restrict__ keys,
                              float* __restrict__ scores) {
    const int b = blockIdx.y, chunk = blockIdx.x;
    __shared__ float qs[HDIM];
    for (int i = threadIdx.x; i < HDIM; i += 256) qs[i] = q[(size_t)b * HDIM + i];
    __syncthreads();
    const int wave = threadIdx.x >> 5, lane = threadIdx.x & 31;
    for (int rr = 0; rr < 16; ++rr) {
        int s = chunk * 128 + wave * 16 + rr;
        const float* krow = keys + ((size_t)b * L + s) * HDIM;
        float sum = 0.f;
#pragma unroll
        for (int c = 0; c < 8; ++c) {
            int h = lane * 4 + c * 128;
            float4 kv = *(const float4*)(krow + h);
            sum += kv.x * qs[h] + kv.y * qs[h + 1] + kv.z * qs[h + 2] + kv.w * qs[h + 3];
        }
#pragma unroll
        for (int off = 16; off; off >>= 1) sum += __shfl_xor(sum, off, 32);
        if (lane == 0) scores[(size_t)b * L + s] = sum * 0.03125f;   // 1/sqrt(1024)
    }
}

// ---------------- K6: softmax -> moments -> r (erf closed form) ----------------
// grid 32 (batch), block 256 (one thread per basis function)
__global__ void finalize_kernel(const float* __restrict__ scores,
                                float* __restrict__ rbuf, float* __restrict__ r_out) {
    const int b = blockIdx.x, tid = threadIdx.x;
    __shared__ float ssc[L];
    __shared__ float red[256];
    for (int i = tid; i < L; i += 256) ssc[i] = scores[(size_t)b * L + i];
    __syncthreads();
    // max
    float lm = -3.4e38f;
#pragma unroll
    for (int c = 0; c < 4; ++c) lm = fmaxf(lm, ssc[tid * 4 + c]);
    red[tid] = lm; __syncthreads();
    for (int off = 128; off; off >>= 1) { if (tid < off) red[tid] = fmaxf(red[tid], red[tid + off]); __syncthreads(); }
    float mx = red[0]; __syncthreads();
    // Z, m1, m2
    float z = 0.f, m1 = 0.f, m2 = 0.f;
#pragma unroll
    for (int c = 0; c < 4; ++c) {
        int s = tid * 4 + c;
        float e = expf(ssc[s] - mx);
        float p = posf(s);
        z += e; m1 += e * p; m2 += e * p * p;
    }
    red[tid] = z; __syncthreads();
    for (int off = 128; off; off >>= 1) { if (tid < off) red[tid] += red[tid + off]; __syncthreads(); }
    float Z = red[0]; __syncthreads();
    red[tid] = m1; __syncthreads();
    for (int off = 128; off; off >>= 1) { if (tid < off) red[tid] += red[tid + off]; __syncthreads(); }
    float M1 = red[0]; __syncthreads();
    red[tid] = m2; __syncthreads();
    for (int off = 128; off; off >>= 1) { if (tid < off) red[tid] += red[tid + off]; __syncthreads(); }
    float M2 = red[0]; __syncthreads();

    float mu  = M1 / Z;
    float ss  = fmaxf(M2 / Z - mu * mu, 1e-6f);
    float th1 = mu / ss, th2 = -0.5f / ss;
    float A   = cbrtf(1.5f * ss);
    float tau = (mu * mu - A * A) / (2.0f * ss);
    float lft = mu - A, rgt = mu + A;

    int j = tid;
    float muj = mu_of(j), sg = sig_of(j);
    float zl = (lft - muj) / sg, zr = (rgt - muj) / sg;
    float phil = expf(-0.5f * zl * zl) * SQRT2PI_INV;
    float phir = expf(-0.5f * zr * zr) * SQRT2PI_INV;
    float Phl = 0.5f * (1.0f + erff(zl * 0.70710678f));
    float Phr = 0.5f * (1.0f + erff(zr * 0.70710678f));
    float dPhi = Phr - Phl, dphi = phil - phir;
    float I0 = dPhi;
    float I1 = muj * dPhi + sg * dphi;
    float I2 = muj * muj * dPhi + 2.0f * muj * sg * dphi + sg * sg * (dPhi - zr * phir + zl * phil);
    float rv = -tau * I0 + th1 * I1 + th2 * I2;
    rbuf[(size_t)b * NB + j]  = rv;
    r_out[(size_t)b * NB + j] = rv;
}

// ---------------- K7: W[b][s] = (G r_b)[s]   M=1024(s), N=32(b), K=256 ----------------
// grid (64,2), block 32
__global__ void gemm_W_wmma(const float* __restrict__ G, const float* __restrict__ R,
                            float* __restrict__ W) {
    const int s0 = blockIdx.x * 16, b0 = blockIdx.y * 16;
    const int lane = threadIdx.x, half = lane >> 4, idx = lane & 15;
    v8f acc = {};
    for (int k = 0; k < NB; k += 4) {
        int kk = k + 2 * half;
        v2f a, b;
        a.x = G[(size_t)(s0 + idx) * NB + kk];   a.y = G[(size_t)(s0 + idx) * NB + kk + 1];
        b.x = R[(size_t)(b0 + idx) * NB + kk];   b.y = R[(size_t)(b0 + idx) * NB + kk + 1];
        acc = __builtin_amdgcn_wmma_f32_16x16x4_f32(false, a, false, b, (short)0, acc, false, false);
    }
#pragma unroll
    for (int t = 0; t < 8; ++t)
        W[(size_t)(b0 + idx) * L + s0 + t + 8 * half] = acc[t];   // W[b][s]
}

// ---------------- K8: partial c: for each (s-chunk, b): sum_s V[b,s,h]*W[b,s] ----------------
// grid (16 chunks of 64 s, 32 batches), block 256, 4 h per thread (float4)
__global__ void ctx_partial(const float* __restrict__ values, const float* __restrict__ W,
                            float* __restrict__ partial) {
    const int b = blockIdx.y, ch = blockIdx.x;
    __shared__ float wsh[64];
    if (threadIdx.x < 64) wsh[threadIdx.x] = W[(size_t)b * L + ch * 64 + threadIdx.x];
    __syncthreads();
    const int h = threadIdx.x * 4;
    const float* vb = values + (size_t)b * L * HDIM + (size_t)ch * 64 * HDIM + h;
    float ax = 0.f, ay = 0.f, az = 0.f, aw = 0.f;
    for (int ss = 0; ss < 64; ++ss) {
        float4 v = *(const float4*)(vb + (size_t)ss * HDIM);
        float wv = wsh[ss];
        ax += v.x * wv; ay += v.y * wv; az += v.z * wv; aw += v.w * wv;
    }
    float4 o; o.x = ax; o.y = ay; o.z = az; o.w = aw;
    *(float4*)(partial + ((size_t)(ch * 32 + b)) * HDIM + h) = o;
}

// ---------------- K9: reduce partials -> c ----------------
__global__ void ctx_reduce(const float* __restrict__ partial, float* __restrict__ c_out) {
    int idx = blockIdx.x * 256 + threadIdx.x;      // 0..32767 == b*1024+h
    int b = idx >> 10, h = idx & 1023;
    float s = 0.f;
#pragma unroll
    for (int ch = 0; ch < 16; ++ch) s += partial[((size_t)(ch * 32 + b)) * HDIM + h];
    c_out[idx] = s;
}

// ---------------- workspace layout (bytes) ----------------
#define OFF_F   ((size_t)0)                  // 256*1024 f32   = 1 MB
#define OFF_AUG ((size_t)1 << 20)            // 256*512  f64   = 1 MB
#define OFF_G   ((size_t)2 << 20)            // 1024*256 f32   = 1 MB
#define OFF_SC  ((size_t)3 << 20)            // 32*1024  f32   = 128 KB
#define OFF_R   (OFF_SC + 131072)            // 32*256   f32   = 32 KB
#define OFF_W   (OFF_R + 32768)              // 32*1024  f32   = 128 KB
#define OFF_P   (OFF_W + 131072)             // 16*32*1024 f32 = 2 MB

extern "C" void kernel_launch(void* const* d_in, const int* in_sizes, int n_in,
                              void* d_out, int out_size, void* d_ws, size_t ws_size,
                              hipStream_t stream) {
    const float* query  = (const float*)d_in[0];
    const float* keys   = (const float*)d_in[1];
    const float* values = (const float*)d_in[2];
    float* out = (float*)d_out;                 // [c (32*1024) | r (32*256)]
    char*  ws  = (char*)d_ws;

    float*  F       = (float*)(ws + OFF_F);
    double* Aug     = (double*)(ws + OFF_AUG);
    float*  G       = (float*)(ws + OFF_G);
    float*  scores  = (float*)(ws + OFF_SC);
    float*  rbuf    = (float*)(ws + OFF_R);
    float*  W       = (float*)(ws + OFF_W);
    float*  partial = (float*)(ws + OFF_P);
    float*  c_out   = out;
    float*  r_out   = out + BS * HDIM;

    // --- precompute G = F^T (F F^T + 0.01 I)^-1 (constant in reference, not an input) ---
    fill_F      <<<dim3(1024), dim3(256), 0, stream>>>(F);
    gemm_M_wmma <<<dim3(16, 16), dim3(32), 0, stream>>>(F, Aug);
    gauss_jordan<<<dim3(1), dim3(1024), 0, stream>>>(Aug);
    gemm_G_wmma <<<dim3(64, 16), dim3(32), 0, stream>>>(F, Aug, G);

    // --- encoder: scores -> softmax moments -> r ---
    scores_kernel  <<<dim3(8, 32), dim3(256), 0, stream>>>(query, keys, scores);
    finalize_kernel<<<dim3(32), dim3(256), 0, stream>>>(scores, rbuf, r_out);

    // --- decoder: c[b,h] = sum_s V[b,s,h] * (G r_b)[s]  (avoids 17-GFLOP V^T G) ---
    gemm_W_wmma<<<dim3(64, 2), dim3(32), 0, stream>>>(G, rbuf, W);
    ctx_partial<<<dim3(16, 32), dim3(256), 0, stream>>>(values, W, partial);
    ctx_reduce <<<dim3(128), dim3(256), 0, stream>>>(partial, c_out);
}